// model_28716151341569
// MI455X (gfx1250) — compile-verified
//
#include <hip/hip_runtime.h>
#include <hip/hip_bf16.h>
#include <stdint.h>

// ---------------------------------------------------------------------------
// Seq2seq (biLSTM encoder + LSTM decoder w/ Luong attention + generator)
// for MI455X / gfx1250.  fp32 throughout, WMMA f32 16x16x4 for all GEMMs.
// Weights are pre-packed into a K-pair layout so every WMMA B-fragment is a
// single coalesced 8-byte load; A-fragments are 8-byte v2f loads from LDS or global memory.
// Async direct-to-LDS staging is used where the toolchain exposes it.
// ---------------------------------------------------------------------------

typedef float v2f __attribute__((ext_vector_type(2)));
typedef float v8f __attribute__((ext_vector_type(8)));

#define SEQ 100
#define TT  40
#define BB  64
#define HSZ 256
#define HH  128
#define VSZ 30000

#if defined(__has_builtin)
# if __has_builtin(__builtin_amdgcn_global_load_async_to_lds_b128) && \
     __has_builtin(__builtin_amdgcn_s_wait_asynccnt)
#  define ASYNC_LDS 1
# endif
#endif
#ifndef ASYNC_LDS
# define ASYNC_LDS 0
#endif

#if ASYNC_LDS
// Builtin signature (from hipcc diagnostic): GCC-style int4 vector pointers,
// global (AS1) source, LDS (AS3) destination, imm offset, imm cpol.
typedef int vi4 __attribute__((vector_size(16)));
typedef __attribute__((address_space(1))) vi4 gvi4;
typedef __attribute__((address_space(3))) vi4 lvi4;
#endif

// 16B global -> LDS stage (async path uses GLOBAL_LOAD_ASYNC_TO_LDS_B128,
// tracked with ASYNCcnt; fallback is a plain float4 round trip).
__device__ __forceinline__ void stage_b128(const float* g, float* l) {
#if ASYNC_LDS
  __builtin_amdgcn_global_load_async_to_lds_b128((gvi4*)g, (lvi4*)l, 0, 0);
#else
  *(float4*)l = *(const float4*)g;
#endif
}
__device__ __forceinline__ void stage_wait() {
#if ASYNC_LDS
  __builtin_amdgcn_s_wait_asynccnt(0);
#endif
}

__device__ __forceinline__ v8f wmma4(v2f a, v2f b, v8f c) {
  // D = A(16x4) * B(4x16) + C(16x16), fp32 exact (matches reference math)
  return __builtin_amdgcn_wmma_f32_16x16x4_f32(false, a, false, b, (short)0, c,
                                               false, false);
}

__device__ __forceinline__ float sigm(float x) { return 1.0f / (1.0f + __expf(-x)); }

// Accumulate 4 M-tiles (rows 0..63) x one 16-col N-tile over K.
// A: row-major activations (lda floats/row).
// Bp: K-pair packed weights: float2 P[k2][n] = (W[n][2k2], W[n][2k2+1]),
//     ldb = N, so each lane's B-fragment is ONE v2f load, contiguous across
//     lanes 0..15 (128B burst per half-wave).
__device__ __forceinline__ void accum_k(const float* __restrict__ A, int lda,
                                        const float* __restrict__ Bp, int ldb,
                                        int col, int K, int mrow16, int k2_off,
                                        v8f& a0, v8f& a1, v8f& a2, v8f& a3) {
#pragma unroll 4
  for (int k = 0; k < K; k += 4) {
    const int kb = k + (k2_off << 1);       // lanes 16..31 take K+2,K+3
    v2f bf = *(const v2f*)(Bp + ((size_t)((k >> 1) + k2_off) * ldb + col) * 2);
    v2f x0 = *(const v2f*)(A + (size_t)(mrow16)      * lda + kb);
    v2f x1 = *(const v2f*)(A + (size_t)(16 + mrow16) * lda + kb);
    v2f x2 = *(const v2f*)(A + (size_t)(32 + mrow16) * lda + kb);
    v2f x3 = *(const v2f*)(A + (size_t)(48 + mrow16) * lda + kb);
    a0 = wmma4(x0, bf, a0);
    a1 = wmma4(x1, bf, a1);
    a2 = wmma4(x2, bf, a2);
    a3 = wmma4(x3, bf, a3);
  }
}

// C/D layout: VGPR v, lanes 0-15 -> M=v, lanes 16-31 -> M=8+v; N = lane%16.
__device__ __forceinline__ void store_tiles(float* dst, int ldd, int col, int hi,
                                            v8f a0, v8f a1, v8f a2, v8f a3) {
#pragma unroll
  for (int v = 0; v < 8; ++v) {
    dst[(size_t)(v + hi) * ldd + col]      = a0[v];
    dst[(size_t)(16 + v + hi) * ldd + col] = a1[v];
    dst[(size_t)(32 + v + hi) * ldd + col] = a2[v];
    dst[(size_t)(48 + v + hi) * ldd + col] = a3[v];
  }
}

// --------------------------- utility kernels -------------------------------

// Pack weight W (N x K, row-major, as in PyTorch W[n][k]) into the K-pair
// layout described above: P[((k>>1)*N + n)*2 + (k&1)] = W[n][k].
__global__ void k_pack_pairs(const float* __restrict__ W, float* __restrict__ P,
                             int N, int K) {
  size_t idx = (size_t)blockIdx.x * 256 + threadIdx.x;
  if (idx >= (size_t)N * K) return;
  int n = (int)(idx / K), k = (int)(idx % K);
  P[((size_t)(k >> 1) * N + n) * 2 + (k & 1)] = W[idx];
}

__global__ void k_embed_enc(const int* __restrict__ inp, const float* __restrict__ emb,
                            float* __restrict__ x) {  // x: (S,B,H)
  size_t idx = (size_t)blockIdx.x * 256 + threadIdx.x;
  if (idx >= (size_t)SEQ * BB * HSZ) return;
  int h = (int)(idx & (HSZ - 1));
  size_t sb = idx >> 8;
  int b = (int)(sb % BB), s = (int)(sb / BB);
  int tok = inp[b * SEQ + s];
  x[idx] = emb[(size_t)tok * HSZ + h];
}

__global__ void k_embed_dec(const int* __restrict__ outTok, const float* __restrict__ emb,
                            float* __restrict__ x) {  // x: (T,B,H), teacher forcing
  size_t idx = (size_t)blockIdx.x * 256 + threadIdx.x;
  if (idx >= (size_t)TT * BB * HSZ) return;
  int h = (int)(idx & (HSZ - 1));
  size_t tb = idx >> 8;
  int b = (int)(tb % BB), t = (int)(tb / BB);
  int tok = (t == 0) ? 3 : outTok[b * TT + (t - 1)];
  x[idx] = emb[(size_t)tok * HSZ + h];
}

// --------------------------- encoder layer ---------------------------------
// grid.x = 2 (fwd/bwd direction); 256 threads = 8 waves. State lives in LDS.
extern "C" __global__ __launch_bounds__(256) void k_enc_layer(
    const float* __restrict__ xin, float* __restrict__ xout,
    const float* __restrict__ WihP_f, const float* __restrict__ WihP_b,
    const float* __restrict__ WhhP_f, const float* __restrict__ WhhP_b,
    const float* __restrict__ bih_f, const float* __restrict__ bhh_f,
    const float* __restrict__ bih_b, const float* __restrict__ bhh_b,
    float* __restrict__ hDec, float* __restrict__ cDec, int layer) {
  extern __shared__ float smem[];
  float* sX = smem;             // 64 x 256 staged input (64 KB)
  float* sH = sX + BB * HSZ;    // 64 x 128 hidden
  float* sC = sH + BB * HH;     // 64 x 128 cell
  float* sG = sC + BB * HH;     // 64 x 512 gates (128 KB)   total 256 KB
  const int tid = threadIdx.x;
  const int dir = blockIdx.x;
  const int wave = tid >> 5, lane = tid & 31;
  const int mrow16 = lane & 15;
  const int k2_off = lane >> 4;           // 0 | 1 (K-pair select)
  const int hi = (lane >> 4) << 3;
  const float* WihP = dir ? WihP_b : WihP_f;   // packed (256, 512)
  const float* WhhP = dir ? WhhP_b : WhhP_f;   // packed (128, 512)
  const float* bih  = dir ? bih_b  : bih_f;
  const float* bhh  = dir ? bhh_b  : bhh_f;

  for (int i = tid; i < BB * HH; i += 256) { sH[i] = 0.0f; sC[i] = 0.0f; }
  __syncthreads();

  for (int step = 0; step < SEQ; ++step) {
    const int t = dir ? (SEQ - 1 - step) : step;
    const float* xt = xin + (size_t)t * BB * HSZ;
    for (int i4 = tid; i4 < (BB * HSZ) / 4; i4 += 256)
      stage_b128(xt + (size_t)i4 * 4, sX + (size_t)i4 * 4);
    stage_wait();
    __syncthreads();

    // gates G = X_t @ Wih^T + H @ Whh^T + b   (64 x 512)
    for (int ntb = 0; ntb < 4; ++ntb) {
      const int nt = wave + ntb * 8;
      const int col = nt * 16 + mrow16;
      const float bias = bih[col] + bhh[col];
      v8f a0, a1, a2, a3;
      for (int v = 0; v < 8; ++v) { a0[v] = bias; a1[v] = bias; a2[v] = bias; a3[v] = bias; }
      accum_k(sX, HSZ, WihP, 4 * HH, col, HSZ, mrow16, k2_off, a0, a1, a2, a3);
      accum_k(sH, HH,  WhhP, 4 * HH, col, HH,  mrow16, k2_off, a0, a1, a2, a3);
      store_tiles(sG, 4 * HH, col, hi, a0, a1, a2, a3);
    }
    __syncthreads();

    // LSTM cell elementwise (i,f,g,o)
    for (int i = tid; i < BB * HH; i += 256) {
      const int b = i >> 7, u = i & (HH - 1);
      const float* g = sG + (size_t)b * (4 * HH);
      float cn = sigm(g[HH + u]) * sC[i] + sigm(g[u]) * tanhf(g[2 * HH + u]);
      float hn = sigm(g[3 * HH + u]) * tanhf(cn);
      sC[i] = cn; sH[i] = hn;
      xout[((size_t)t * BB + b) * HSZ + dir * HH + u] = hn;
    }
    __syncthreads();
  }
  // final states feed decoder h0/c0 (concat fwd|bwd)
  for (int i = tid; i < BB * HH; i += 256) {
    const int b = i >> 7, u = i & (HH - 1);
    hDec[((size_t)layer * BB + b) * HSZ + dir * HH + u] = sH[i];
    cDec[((size_t)layer * BB + b) * HSZ + dir * HH + u] = sC[i];
  }
}

// --------------------------- decoder (persistent) --------------------------
// One workgroup, 8 waves, 40 sequential steps. Gates in 256 KB LDS; h/c/op in
// global workspace. Logits are deferred to k_logits (teacher forcing).
extern "C" __global__ __launch_bounds__(256) void k_decoder(
    const float* __restrict__ demb,   // (T,B,256)
    float* __restrict__ h,            // (2,B,256)  init = encoder finals
    float* __restrict__ c,            // (2,B,256)
    float* __restrict__ opPrev,       // (B,256)
    float* __restrict__ opAll,        // (T,B,256)
    const float* __restrict__ enc,    // (S,B,256)
    const float* __restrict__ Wih0P,  // packed (512,1024)
    const float* __restrict__ Wih1P,  // packed (256,1024)
    const float* __restrict__ Whh0P,  // packed (256,1024)
    const float* __restrict__ Whh1P,  // packed (256,1024)
    const float* __restrict__ bih0, const float* __restrict__ bhh0,
    const float* __restrict__ bih1, const float* __restrict__ bhh1,
    const float* __restrict__ lininP,   // packed (256,256)
    const float* __restrict__ linoutP)  // packed (512,256)
{
  extern __shared__ float smem[];
  float* sG  = smem;                 // 64x1024 gates (256 KB), reused below:
  float* sQ  = smem;                 // 64x256 attention query
  float* sW  = smem + BB * HSZ;      // 64x100 scores -> softmax weights
  float* sCC = sW + BB * SEQ;        // 64x256 context
  const int tid = threadIdx.x;
  const int wave = tid >> 5, lane = tid & 31;
  const int mrow16 = lane & 15, k2_off = lane >> 4, hi = (lane >> 4) << 3;
  float* h0 = h;            float* h1 = h + BB * HSZ;
  float* c0 = c;            float* c1 = c + BB * HSZ;

  for (int i = tid; i < BB * HSZ; i += 256) opPrev[i] = 0.0f;
  __threadfence_block();
  __syncthreads();

  for (int t = 0; t < TT; ++t) {
    const float* xt = demb + (size_t)t * BB * HSZ;

    // ---- layer 0 gates: [demb, opPrev] @ Wih0^T + h0 @ Whh0^T + b ----
    for (int ntb = 0; ntb < 8; ++ntb) {
      const int col = (wave + ntb * 8) * 16 + mrow16;
      const float bias = bih0[col] + bhh0[col];
      v8f a0, a1, a2, a3;
      for (int v = 0; v < 8; ++v) { a0[v] = bias; a1[v] = bias; a2[v] = bias; a3[v] = bias; }
      accum_k(xt,     HSZ, Wih0P,                            4 * HSZ, col, HSZ, mrow16, k2_off, a0, a1, a2, a3);
      accum_k(opPrev, HSZ, Wih0P + (size_t)HSZ * (4 * HSZ),  4 * HSZ, col, HSZ, mrow16, k2_off, a0, a1, a2, a3);
      accum_k(h0,     HSZ, Whh0P,                            4 * HSZ, col, HSZ, mrow16, k2_off, a0, a1, a2, a3);
      store_tiles(sG, 4 * HSZ, col, hi, a0, a1, a2, a3);
    }
    __syncthreads();
    for (int i = tid; i < BB * HSZ; i += 256) {
      const int b = i >> 8, u = i & (HSZ - 1);
      const float* g = sG + (size_t)b * (4 * HSZ);
      float cn = sigm(g[HSZ + u]) * c0[i] + sigm(g[u]) * tanhf(g[2 * HSZ + u]);
      float hn = sigm(g[3 * HSZ + u]) * tanhf(cn);
      c0[i] = cn; h0[i] = hn;
    }
    __threadfence_block();
    __syncthreads();

    // ---- layer 1 gates: h0 @ Wih1^T + h1 @ Whh1^T + b ----
    for (int ntb = 0; ntb < 8; ++ntb) {
      const int col = (wave + ntb * 8) * 16 + mrow16;
      const float bias = bih1[col] + bhh1[col];
      v8f a0, a1, a2, a3;
      for (int v = 0; v < 8; ++v) { a0[v] = bias; a1[v] = bias; a2[v] = bias; a3[v] = bias; }
      accum_k(h0, HSZ, Wih1P, 4 * HSZ, col, HSZ, mrow16, k2_off, a0, a1, a2, a3);
      accum_k(h1, HSZ, Whh1P, 4 * HSZ, col, HSZ, mrow16, k2_off, a0, a1, a2, a3);
      store_tiles(sG, 4 * HSZ, col, hi, a0, a1, a2, a3);
    }
    __syncthreads();
    for (int i = tid; i < BB * HSZ; i += 256) {
      const int b = i >> 8, u = i & (HSZ - 1);
      const float* g = sG + (size_t)b * (4 * HSZ);
      float cn = sigm(g[HSZ + u]) * c1[i] + sigm(g[u]) * tanhf(g[2 * HSZ + u]);
      float hn = sigm(g[3 * HSZ + u]) * tanhf(cn);
      c1[i] = cn; h1[i] = hn;
    }
    __threadfence_block();
    __syncthreads();

    // ---- attention: q = h1 @ W_linin^T ----
    for (int ntb = 0; ntb < 2; ++ntb) {
      const int col = (wave + ntb * 8) * 16 + mrow16;
      v8f a0, a1, a2, a3;
      for (int v = 0; v < 8; ++v) { a0[v] = 0.f; a1[v] = 0.f; a2[v] = 0.f; a3[v] = 0.f; }
      accum_k(h1, HSZ, lininP, HSZ, col, HSZ, mrow16, k2_off, a0, a1, a2, a3);
      store_tiles(sQ, HSZ, col, hi, a0, a1, a2, a3);
    }
    __syncthreads();

    // scores[b][s] = enc[b,s,:] . q[b,:]
    for (int i = tid; i < BB * SEQ; i += 256) {
      const int b = i / SEQ, s = i % SEQ;
      const float* e = enc + ((size_t)s * BB + b) * HSZ;
      const float* q = sQ + (size_t)b * HSZ;
      float acc = 0.f;
      for (int k = 0; k < HSZ; k += 4) {
        float4 ev = *(const float4*)(e + k);
        acc += ev.x * q[k] + ev.y * q[k + 1] + ev.z * q[k + 2] + ev.w * q[k + 3];
      }
      sW[i] = acc;
    }
    __syncthreads();
    if (tid < BB) {                      // softmax over S per batch row
      float* row = sW + tid * SEQ;
      float m = row[0];
      for (int s = 1; s < SEQ; ++s) m = fmaxf(m, row[s]);
      float sum = 0.f;
      for (int s = 0; s < SEQ; ++s) { float e = __expf(row[s] - m); row[s] = e; sum += e; }
      float inv = 1.0f / sum;
      for (int s = 0; s < SEQ; ++s) row[s] *= inv;
    }
    __syncthreads();
    // context cc[b][u] = sum_s w[b][s] * enc[b,s,u]
    for (int i = tid; i < BB * HSZ; i += 256) {
      const int b = i >> 8, u = i & (HSZ - 1);
      const float* wrow = sW + (size_t)b * SEQ;
      float acc = 0.f;
      for (int s = 0; s < SEQ; ++s)
        acc += wrow[s] * enc[((size_t)s * BB + b) * HSZ + u];
      sCC[i] = acc;
    }
    __syncthreads();

    // op = tanh([cc, h1] @ W_linout^T); store to opPrev and opAll[t]
    for (int ntb = 0; ntb < 2; ++ntb) {
      const int col = (wave + ntb * 8) * 16 + mrow16;
      v8f a0, a1, a2, a3;
      for (int v = 0; v < 8; ++v) { a0[v] = 0.f; a1[v] = 0.f; a2[v] = 0.f; a3[v] = 0.f; }
      accum_k(sCC, HSZ, linoutP,                     HSZ, col, HSZ, mrow16, k2_off, a0, a1, a2, a3);
      accum_k(h1,  HSZ, linoutP + (size_t)HSZ * HSZ, HSZ, col, HSZ, mrow16, k2_off, a0, a1, a2, a3);
#pragma unroll
      for (int v = 0; v < 8; ++v) {
        float o0 = tanhf(a0[v]); int r0 = v + hi;
        float o1 = tanhf(a1[v]); int r1 = 16 + v + hi;
        float o2 = tanhf(a2[v]); int r2 = 32 + v + hi;
        float o3 = tanhf(a3[v]); int r3 = 48 + v + hi;
        opPrev[(size_t)r0 * HSZ + col] = o0; opAll[((size_t)t * BB + r0) * HSZ + col] = o0;
        opPrev[(size_t)r1 * HSZ + col] = o1; opAll[((size_t)t * BB + r1) * HSZ + col] = o1;
        opPrev[(size_t)r2 * HSZ + col] = o2; opAll[((size_t)t * BB + r2) * HSZ + col] = o2;
        opPrev[(size_t)r3 * HSZ + col] = o3; opAll[((size_t)t * BB + r3) * HSZ + col] = o3;
      }
    }
    __threadfence_block();
    __syncthreads();
  }
}

// --------------------------- logits GEMM -----------------------------------
// (T*B=2560, 256) @ (256, 30000) + b_gen.  grid = (160 M-tiles, 375), 5 waves
// per block each owning one 16-col N-tile; A tile staged (async) into LDS,
// packed W_gen streamed from L2 with global_prefetch.
extern "C" __global__ __launch_bounds__(160) void k_logits(
    const float* __restrict__ OP,    // (2560,256), row = t*B + b
    const float* __restrict__ WgP,   // packed (256,30000)
    const float* __restrict__ bgen,  // (30000)
    float* __restrict__ outp) {      // (B,T,VSZ)
  __shared__ float sA[16 * HSZ];
  const int tid = threadIdx.x;
  const int wave = tid >> 5, lane = tid & 31;
  const int mrow16 = lane & 15, k2_off = lane >> 4, hi = (lane >> 4) << 3;
  const int mt = blockIdx.x;
  const int nt = blockIdx.y * 5 + wave;
  const int col = nt * 16 + mrow16;

  const float* src = OP + (size_t)mt * 16 * HSZ;
  for (int i4 = tid; i4 < (16 * HSZ) / 4; i4 += 160)
    stage_b128(src + (size_t)i4 * 4, sA + (size_t)i4 * 4);
  stage_wait();
  __syncthreads();

  const float bias = bgen[col];
  v8f acc;
  for (int v = 0; v < 8; ++v) acc[v] = bias;
#pragma unroll 4
  for (int k = 0; k < HSZ; k += 4) {
    if ((k & 63) == 0 && (k + 64) < HSZ)
      __builtin_prefetch(WgP + ((size_t)(((k + 64) >> 1) + k2_off) * VSZ + col) * 2, 0, 1);
    v2f bf = *(const v2f*)(WgP + ((size_t)((k >> 1) + k2_off) * VSZ + col) * 2);
    v2f av = *(const v2f*)(sA + (size_t)mrow16 * HSZ + k + (k2_off << 1));
    acc = wmma4(av, bf, acc);
  }
#pragma unroll
  for (int v = 0; v < 8; ++v) {
    const int r = mt * 16 + v + hi;       // r = t*B + b
    const int t = r >> 6, b = r & (BB - 1);
    outp[((size_t)b * TT + t) * VSZ + col] = acc[v];
  }
}

// --------------------------- host launcher ---------------------------------

extern "C" void kernel_launch(void* const* d_in, const int* in_sizes, int n_in,
                              void* d_out, int out_size, void* d_ws, size_t ws_size,
                              hipStream_t stream) {
  const int* inp     = (const int*)d_in[0];
  const int* outTok  = (const int*)d_in[1];
  const float* encemb = (const float*)d_in[2];
  const float* decemb = (const float*)d_in[3];
  const float* eW[4][4];   // [layer*2+dir][Wih,Whh,bih,bhh]
  for (int i = 0; i < 4; ++i)
    for (int j = 0; j < 4; ++j) eW[i][j] = (const float*)d_in[4 + i * 4 + j];
  const float* dW[2][4];   // [layer][Wih,Whh,bih,bhh]
  for (int l = 0; l < 2; ++l)
    for (int j = 0; j < 4; ++j) dW[l][j] = (const float*)d_in[20 + l * 4 + j];
  const float* W_linin  = (const float*)d_in[28];
  const float* W_linout = (const float*)d_in[29];
  const float* W_gen    = (const float*)d_in[30];
  const float* b_gen    = (const float*)d_in[31];

  float* ws = (float*)d_ws;
  size_t o = 0;
  auto take = [&](size_t n) { float* p = ws + o; o += n; return p; };
  float* encWihP[4]; for (int i = 0; i < 4; ++i) encWihP[i] = take(256 * 512);
  float* encWhhP[4]; for (int i = 0; i < 4; ++i) encWhhP[i] = take(128 * 512);
  float* dWih0P = take(512 * 1024);
  float* dWih1P = take(256 * 1024);
  float* dWhhP[2]; for (int l = 0; l < 2; ++l) dWhhP[l] = take(256 * 1024);
  float* lininP  = take(256 * 256);
  float* linoutP = take(512 * 256);
  float* WgP     = take((size_t)256 * VSZ);
  float* xA      = take((size_t)SEQ * BB * HSZ);
  float* xB      = take((size_t)SEQ * BB * HSZ);
  float* demb    = take((size_t)TT * BB * HSZ);
  float* opAll   = take((size_t)TT * BB * HSZ);
  float* hDec    = take((size_t)2 * BB * HSZ);
  float* cDec    = take((size_t)2 * BB * HSZ);
  float* opPrev  = take((size_t)BB * HSZ);

  auto pack = [&](const float* s, float* d, int N, int K) {
    int total = N * K;
    k_pack_pairs<<<(total + 255) / 256, 256, 0, stream>>>(s, d, N, K);
  };
  for (int i = 0; i < 4; ++i) pack(eW[i][0], encWihP[i], 512, 256);
  for (int i = 0; i < 4; ++i) pack(eW[i][1], encWhhP[i], 512, 128);
  pack(dW[0][0], dWih0P, 1024, 512);
  pack(dW[1][0], dWih1P, 1024, 256);
  pack(dW[0][1], dWhhP[0], 1024, 256);
  pack(dW[1][1], dWhhP[1], 1024, 256);
  pack(W_linin,  lininP,  256, 256);
  pack(W_linout, linoutP, 256, 512);
  pack(W_gen,    WgP,     VSZ, 256);

  k_embed_enc<<<(SEQ * BB * HSZ + 255) / 256, 256, 0, stream>>>(inp, encemb, xA);
  k_embed_dec<<<(TT * BB * HSZ + 255) / 256, 256, 0, stream>>>(outTok, decemb, demb);

  const size_t encLds = (size_t)(BB * HSZ + 2 * BB * HH + BB * 4 * HH) * sizeof(float); // 256 KB
  k_enc_layer<<<2, 256, encLds, stream>>>(xA, xB,
      encWihP[0], encWihP[1], encWhhP[0], encWhhP[1],
      eW[0][2], eW[0][3], eW[1][2], eW[1][3], hDec, cDec, 0);
  k_enc_layer<<<2, 256, encLds, stream>>>(xB, xA,
      encWihP[2], encWihP[3], encWhhP[2], encWhhP[3],
      eW[2][2], eW[2][3], eW[3][2], eW[3][3], hDec, cDec, 1);

  const size_t decLds = (size_t)(BB * 4 * HSZ) * sizeof(float);  // 256 KB
  k_decoder<<<1, 256, decLds, stream>>>(demb, hDec, cDec, opPrev, opAll, xA,
      dWih0P, dWih1P, dWhhP[0], dWhhP[1],
      dW[0][2], dW[0][3], dW[1][2], dW[1][3], lininP, linoutP);

  dim3 lgGrid(160, 375);   // 160 M-tiles x (375 blocks * 5 waves = 1875 N-tiles)
  k_logits<<<lgGrid, 160, 0, stream>>>(opAll, WgP, b_gen, (float*)d_out);
}